// CELoss_Marginal_Smooth_5540507811956
// MI455X (gfx1250) — compile-verified
//
#include <hip/hip_runtime.h>
#include <stdint.h>

#ifndef __has_builtin
#define __has_builtin(x) 0
#endif

#if __has_builtin(__builtin_amdgcn_global_load_async_to_lds_b128)
#define HAVE_ASYNC_LDS 1
#else
#define HAVE_ASYNC_LDS 0
#endif

#if __has_builtin(__builtin_amdgcn_s_wait_asynccnt)
#define WAIT_ASYNCCNT(n) __builtin_amdgcn_s_wait_asynccnt(n)
#else
#define WAIT_ASYNCCNT(n) asm volatile("s_wait_asynccnt %0" ::"i"(n) : "memory")
#endif

typedef int v4i __attribute__((ext_vector_type(4)));
typedef __attribute__((address_space(1))) v4i* as1_v4i_p;
typedef __attribute__((address_space(3))) v4i* as3_v4i_p;

#define NCLASS 12
#define TILE 256          // rows per tile
#define TB 256            // threads per block (8 wave32s)
#define ROW_BYTES (NCLASS * 4)

// Per-row fused loss term.
// smoothed-CE identity: alpha*lpt + (1-alpha)*(att*(row_sum-lpt)+lpt)
//                     = lpt + (1-alpha)*att*(row_sum-lpt)
__device__ __forceinline__ float row_term(const float x[NCLASS], int tt) {
  float m = x[0], sumx = x[0];
#pragma unroll
  for (int j = 1; j < NCLASS; ++j) {
    m = fmaxf(m, x[j]);
    sumx += x[j];
  }
  float s = 0.f;
#pragma unroll
  for (int j = 0; j < NCLASS; ++j) s += __expf(x[j] - m);
  float lse = m + __logf(s);
  float xt = x[0];
#pragma unroll
  for (int j = 1; j < NCLASS; ++j) xt = (tt == j) ? x[j] : xt;
  // attention weight: 1 / (#8-connected neighbors) in a 3x4 grid
  int rr = tt >> 2, cc = tt & 3;
  float nv = (rr == 1) ? 2.f : 1.f;                // vertical neighbors
  float nh = (cc == 0 || cc == 3) ? 1.f : 2.f;     // horizontal neighbors
  float att = 1.f / (nv + nh + nv * nh);           // + diagonals = nv*nh
  float lpt = xt - lse;
  float row_sum = sumx - 12.f * lse;
  return lpt + 0.4f * att * (row_sum - lpt);       // (1-ALPHA) = 0.4
}

__device__ __forceinline__ void unpack12(float4 a, float4 b, float4 c,
                                         float x[NCLASS]) {
  x[0] = a.x;  x[1] = a.y;  x[2] = a.z;  x[3] = a.w;
  x[4] = b.x;  x[5] = b.y;  x[6] = b.z;  x[7] = b.w;
  x[8] = c.x;  x[9] = c.y;  x[10] = c.z; x[11] = c.w;
}

// Deterministic in-block reduction: wave32 shfl tree + LDS across 8 waves.
__device__ __forceinline__ float block_reduce(float v, float* red) {
#pragma unroll
  for (int o = 16; o > 0; o >>= 1) v += __shfl_down(v, o, 32);
  int lane = threadIdx.x & 31, wid = threadIdx.x >> 5;
  if (lane == 0) red[wid] = v;
  __syncthreads();
  float r = 0.f;
  if (threadIdx.x == 0) {
#pragma unroll
    for (int w = 0; w < TB / 32; ++w) r += red[w];
  }
  return r;
}

__global__ void __launch_bounds__(TB)
k_partial(const float* __restrict__ outputs, const int* __restrict__ targets,
          long long ntiles, float* __restrict__ partials) {
  __shared__ float buf[2][TILE * NCLASS];  // 2 x 12KB double buffer
  __shared__ float red[TB / 32];
  const int thr = threadIdx.x;
  float acc = 0.f;

#if HAVE_ASYNC_LDS
  const long long t0 = blockIdx.x;
  const long long tstride = gridDim.x;
  if (t0 < ntiles) {  // prefetch first tile into buf[0]
    const char* g = (const char*)outputs + (size_t)t0 * (TILE * ROW_BYTES);
#pragma unroll
    for (int k = 0; k < 3; ++k) {
      size_t off = ((size_t)thr + (size_t)k * TB) * 16;
      __builtin_amdgcn_global_load_async_to_lds_b128(
          (as1_v4i_p)(g + off),
          (as3_v4i_p)((char*)&buf[0][0] + off), 0, 0);
    }
  }
  int pb = 0;
  for (long long t = t0; t < ntiles; t += tstride, pb ^= 1) {
    long long tn = t + tstride;
    if (tn < ntiles) {  // prefetch next tile into the other buffer
      const char* g = (const char*)outputs + (size_t)tn * (TILE * ROW_BYTES);
#pragma unroll
      for (int k = 0; k < 3; ++k) {
        size_t off = ((size_t)thr + (size_t)k * TB) * 16;
        __builtin_amdgcn_global_load_async_to_lds_b128(
            (as1_v4i_p)(g + off),
            (as3_v4i_p)((char*)&buf[pb ^ 1][0] + off), 0, 0);
      }
      WAIT_ASYNCCNT(3);  // current tile's 3 loads done; next 3 in flight
    } else {
      WAIT_ASYNCCNT(0);
    }
    __syncthreads();
    long long r = t * TILE + thr;
    const float4* rowp =
        reinterpret_cast<const float4*>(&buf[pb][thr * NCLASS]);
    float x[NCLASS];
    unpack12(rowp[0], rowp[1], rowp[2], x);
    acc += row_term(x, targets[r]);
    __syncthreads();  // everyone done reading buf[pb] before it is reloaded
  }
#else
  // Fallback: direct b128 streaming loads, grid-stride over rows.
  const long long nrows = ntiles * TILE;
  for (long long r = (long long)blockIdx.x * TB + thr; r < nrows;
       r += (long long)gridDim.x * TB) {
    const float4* rowp =
        reinterpret_cast<const float4*>(outputs + (size_t)r * NCLASS);
    float x[NCLASS];
    unpack12(rowp[0], rowp[1], rowp[2], x);
    acc += row_term(x, targets[r]);
  }
#endif
  float b = block_reduce(acc, red);
  if (thr == 0) partials[blockIdx.x] = b;
}

__global__ void __launch_bounds__(TB)
k_final(const float* __restrict__ partials, int nparts,
        const float* __restrict__ outputs, const int* __restrict__ targets,
        long long rowBeg, long long rowEnd, float invB,
        float* __restrict__ out) {
  __shared__ float red[TB / 32];
  float acc = 0.f;
  for (int i = threadIdx.x; i < nparts; i += TB) acc += partials[i];
  for (long long r = rowBeg + threadIdx.x; r < rowEnd; r += TB) {
    const float4* rowp =
        reinterpret_cast<const float4*>(outputs + (size_t)r * NCLASS);
    float x[NCLASS];
    unpack12(rowp[0], rowp[1], rowp[2], x);
    acc += row_term(x, targets[r]);
  }
  float tot = block_reduce(acc, red);
  if (threadIdx.x == 0) out[0] = -tot * invB;
}

extern "C" void kernel_launch(void* const* d_in, const int* in_sizes, int n_in,
                              void* d_out, int out_size, void* d_ws,
                              size_t ws_size, hipStream_t stream) {
  const float* outputs = (const float*)d_in[0];
  const int* targets = (const int*)d_in[1];
  const long long B = (long long)in_sizes[1];  // #rows
  const long long ntiles = B / TILE;
  const long long Bfull = ntiles * TILE;
  float* partials = (float*)d_ws;

  int grid = (int)(ntiles < 2048 ? (ntiles > 0 ? ntiles : 1) : 2048);
  k_partial<<<grid, TB, 0, stream>>>(outputs, targets, ntiles, partials);
  k_final<<<1, TB, 0, stream>>>(partials, grid, outputs, targets, Bfull, B,
                                1.0f / (float)B, (float*)d_out);
}